// SSGNNGraphEncoder_71433896067562
// MI455X (gfx1250) — compile-verified
//
#include <hip/hip_runtime.h>

// CDNA5 (gfx1250) wave32 WMMA GNN encoder.
// All dense math uses V_WMMA_F32_16X16X4_F32 (fp32 in/out: exact vs reference).

typedef __attribute__((ext_vector_type(2))) float v2f;
typedef __attribute__((ext_vector_type(8))) float v8f;

__device__ __forceinline__ v8f wmma4(v2f a, v2f b, v8f c) {
  // 8 args: (neg_a, A, neg_b, B, c_mod, C, reuse_a, reuse_b)
  return __builtin_amdgcn_wmma_f32_16x16x4_f32(false, a, false, b, (short)0, c,
                                               false, false);
}

__device__ __forceinline__ float relu_f(float v) { return fmaxf(v, 0.0f); }

// ---------------------------------------------------------------------------
// Generic GEMM: Y[M,64] = act((X [+ Xadd]) @ W[K,64] + bias).
// K, residual-add and ReLU are compile-time -> fully unrolled branch-free
// K/4 x 4 = 64 WMMAs per wave (K=64). One wave -> 16 rows x 64 cols.
// Block = 8 waves = 128 rows. In-place (Y == X) is safe: a wave reads only
// its own 16 rows and stores only at the end.
// ---------------------------------------------------------------------------
template <int K, bool HASADD, bool RELU>
__global__ void __launch_bounds__(256) gemm_bias_act(
    const float* __restrict__ X, const float* __restrict__ Xadd,
    const float* __restrict__ W, const float* __restrict__ bias,
    float* __restrict__ Y, int M) {
  __shared__ float sW[K * 64];
  __shared__ float sB[64];
  const int tid = threadIdx.x;
  for (int i = tid; i < K * 64; i += 256) sW[i] = W[i];
  if (tid < 64) sB[tid] = bias[tid];
  __syncthreads();

  const int wave = tid >> 5;
  const int lane = tid & 31;
  const int rowBase = (blockIdx.x * 8 + wave) << 4;
  if (rowBase >= M) return;

  // A fragment addressing (32-bit A 16x4): lanes 0-15 -> K0,K1; 16-31 -> K2,K3
  const int mA   = lane & 15;
  const int koff = (lane >> 4) << 1;  // 0 or 2
  const int rowA = min(rowBase + mA, M - 1);
  const int nB   = lane & 15;
  const float* __restrict__ xRow = X + (size_t)rowA * K;
  const float* __restrict__ aRow = HASADD ? (Xadd + (size_t)rowA * K) : nullptr;

  const v8f vz = {0.f, 0.f, 0.f, 0.f, 0.f, 0.f, 0.f, 0.f};
  v8f acc[4] = {vz, vz, vz, vz};

#pragma unroll
  for (int kb = 0; kb < (K >> 2); ++kb) {
    const int k0 = (kb << 2) + koff;
    v2f a;
    a.x = xRow[k0];
    a.y = xRow[k0 + 1];
    if (HASADD) {
      a.x += aRow[k0];
      a.y += aRow[k0 + 1];
    }
#pragma unroll
    for (int nt = 0; nt < 4; ++nt) {
      v2f b;
      b.x = sW[k0 * 64 + nt * 16 + nB];
      b.y = sW[(k0 + 1) * 64 + nt * 16 + nB];
      acc[nt] = wmma4(a, b, acc[nt]);
    }
  }

  // C/D layout: VGPR r -> row (r + (lane>>4)*8), col (lane&15) within tile
  const int mBase = rowBase + ((lane >> 4) << 3);
  const bool fullTile = (rowBase + 16) <= M;
#pragma unroll
  for (int nt = 0; nt < 4; ++nt) {
    const int col = nt * 16 + nB;
    const float bv = sB[col];
#pragma unroll
    for (int r = 0; r < 8; ++r) {
      const int row = mBase + r;
      if (fullTile || row < M) {
        float v = acc[nt][r] + bv;
        if (RELU) v = relu_f(v);
        Y[(size_t)row * 64 + col] = v;
      }
    }
  }
}

// ---------------------------------------------------------------------------
// Message passing (one layer): per wave, 16 edges.
//   e(16x64) = edge_attr(16x16) @ W_edge(16x64) + b_edge   (16 WMMAs)
//   agg[dst] += relu(h[src] + e);  agg[src] += relu(h[dst] + e)
// Gather/scatter performed directly in WMMA C-layout; h rows are prefetched
// (global_prefetch_b8) before the WMMA block so L2 gathers overlap the math.
// ---------------------------------------------------------------------------
__global__ void __launch_bounds__(256) message_pass(
    const float* __restrict__ h, const float* __restrict__ edge_attr,
    const int* __restrict__ src, const int* __restrict__ dst,
    const float* __restrict__ W_edge, const float* __restrict__ b_edge,
    float* __restrict__ agg, int E) {
  __shared__ float sW[16 * 64];
  __shared__ float sB[64];
  const int tid = threadIdx.x;
  for (int i = tid; i < 16 * 64; i += 256) sW[i] = W_edge[i];
  if (tid < 64) sB[tid] = b_edge[tid];
  __syncthreads();

  const int wave = tid >> 5;
  const int lane = tid & 31;
  const int eb = (blockIdx.x * 8 + wave) << 4;
  if (eb >= E) return;

  const int mA   = lane & 15;
  const int koff = (lane >> 4) << 1;
  const int nB   = lane & 15;

  // Edge endpoints this lane owns in the C layout -- load early, prefetch the
  // h rows we will gather after the WMMA block.
  const int mOff = (lane >> 4) << 3;
  int rs[8], rd[8];
#pragma unroll
  for (int r = 0; r < 8; ++r) {
    rs[r] = src[eb + mOff + r];
    rd[r] = dst[eb + mOff + r];
  }
#pragma unroll
  for (int r = 0; r < 8; ++r) {
    __builtin_prefetch(&h[(size_t)rs[r] * 64 + nB], 0, 1);
    __builtin_prefetch(&h[(size_t)rd[r] * 64 + nB], 0, 1);
  }

  const float* __restrict__ eaRow = edge_attr + (size_t)(eb + mA) * 16;

  const v8f vz = {0.f, 0.f, 0.f, 0.f, 0.f, 0.f, 0.f, 0.f};
  v8f acc[4] = {vz, vz, vz, vz};

#pragma unroll
  for (int kb = 0; kb < 4; ++kb) {
    const int k0 = (kb << 2) + koff;
    v2f a;
    a.x = eaRow[k0];
    a.y = eaRow[k0 + 1];
#pragma unroll
    for (int nt = 0; nt < 4; ++nt) {
      v2f b;
      b.x = sW[k0 * 64 + nt * 16 + nB];
      b.y = sW[(k0 + 1) * 64 + nt * 16 + nB];
      acc[nt] = wmma4(a, b, acc[nt]);
    }
  }

#pragma unroll
  for (int nt = 0; nt < 4; ++nt) {
    const int col = nt * 16 + nB;
    const float bv = sB[col];
#pragma unroll
    for (int r = 0; r < 8; ++r) {
      const float ev = acc[nt][r] + bv;
      const float hs = h[(size_t)rs[r] * 64 + col];
      const float hd = h[(size_t)rd[r] * 64 + col];
      unsafeAtomicAdd(&agg[(size_t)rd[r] * 64 + col], relu_f(hs + ev));
      unsafeAtomicAdd(&agg[(size_t)rs[r] * 64 + col], relu_f(hd + ev));
    }
  }
}

// ---------------------------------------------------------------------------
__global__ void zero_f32(float* __restrict__ p, int n) {
  int i = blockIdx.x * blockDim.x + threadIdx.x;
  if (i < n) p[i] = 0.0f;
}

__global__ void pool_sum(const float* __restrict__ h, const int* __restrict__ batch,
                         float* __restrict__ gsum, float* __restrict__ counts, int N) {
  const int node = blockIdx.x * 4 + (threadIdx.x >> 6);
  const int ch = threadIdx.x & 63;
  if (node < N) {
    const int b = batch[node];
    unsafeAtomicAdd(&gsum[(size_t)b * 64 + ch], h[(size_t)node * 64 + ch]);
    if (ch == 0) unsafeAtomicAdd(&counts[b], 1.0f);
  }
}

__global__ void pool_finalize(const float* __restrict__ gsum,
                              const float* __restrict__ counts,
                              float* __restrict__ g, int total) {
  int i = blockIdx.x * blockDim.x + threadIdx.x;
  if (i < total) g[i] = gsum[i] / fmaxf(counts[i >> 6], 1.0f);
}

// ---------------------------------------------------------------------------
extern "C" void kernel_launch(void* const* d_in, const int* in_sizes, int n_in,
                              void* d_out, int out_size, void* d_ws, size_t ws_size,
                              hipStream_t stream) {
  const float* x         = (const float*)d_in[0];
  const float* edge_attr = (const float*)d_in[1];
  const int*   edge_idx  = (const int*)d_in[2];
  const int*   batch     = (const int*)d_in[3];
  const float* W_in   = (const float*)d_in[4];
  const float* b_in   = (const float*)d_in[5];
  const float* W_edge = (const float*)d_in[6];
  const float* b_edge = (const float*)d_in[7];
  const float* W_mlp  = (const float*)d_in[8];
  const float* b_mlp  = (const float*)d_in[9];
  const float* W_out  = (const float*)d_in[10];
  const float* b_out  = (const float*)d_in[11];
  float* out = (float*)d_out;

  const int N = in_sizes[0] / 64;   // 50000
  const int E = in_sizes[2] / 2;    // 800000
  const int G = out_size / 64;      // 64 graphs

  // workspace layout
  float* h      = (float*)d_ws;
  float* agg    = h + (size_t)N * 64;
  float* gsum   = agg + (size_t)N * 64;
  float* counts = gsum + (size_t)G * 64;
  float* g      = counts + G;

  const int gemmBlocks = (N + 127) / 128;
  const int msgBlocks  = (E + 127) / 128;
  const int* src = edge_idx;
  const int* dst = edge_idx + E;

  // h = x @ W_in + b_in
  gemm_bias_act<64, false, false><<<gemmBlocks, 256, 0, stream>>>(
      x, nullptr, W_in, b_in, h, N);

  for (int l = 0; l < 3; ++l) {
    zero_f32<<<(N * 64 + 255) / 256, 256, 0, stream>>>(agg, N * 64);
    message_pass<<<msgBlocks, 256, 0, stream>>>(h, edge_attr, src, dst,
                                                W_edge, b_edge, agg, E);
    // h = relu((h + agg) @ W_mlp[l,0] + b)   (residual fused into A-load)
    gemm_bias_act<64, true, true><<<gemmBlocks, 256, 0, stream>>>(
        h, agg, W_mlp + (size_t)(l * 2 + 0) * 64 * 64, b_mlp + (l * 2 + 0) * 64,
        h, N);
    // h = relu(h @ W_mlp[l,1] + b)
    gemm_bias_act<64, false, true><<<gemmBlocks, 256, 0, stream>>>(
        h, nullptr, W_mlp + (size_t)(l * 2 + 1) * 64 * 64, b_mlp + (l * 2 + 1) * 64,
        h, N);
  }

  // mean pooling per graph
  zero_f32<<<(G * 129 + 255) / 256, 256, 0, stream>>>(gsum, G * 129);  // gsum+counts+g
  pool_sum<<<(N + 3) / 4, 256, 0, stream>>>(h, batch, gsum, counts, N);
  pool_finalize<<<(G * 64 + 255) / 256, 256, 0, stream>>>(gsum, counts, g, G * 64);

  // out = g @ W_out + b_out
  gemm_bias_act<64, false, false><<<(G + 127) / 128, 256, 0, stream>>>(
      g, nullptr, W_out, b_out, out, G);
}